// CubicModelLarge_16587163697192
// MI455X (gfx1250) — compile-verified
//
#include <hip/hip_runtime.h>
#include <stddef.h>

// Cubic-expansion MLP: y = phi(x) @ W^T + b, phi = [x | triu(x⊗x) | x ⊗ triu(x⊗x)]
// GEMM M=1024, N=64/10, K=135264 with phi generated on the fly from LDS tiles.
// fp32 path -> V_WMMA_F32_16X16X4_F32 (keeps reference fp32 precision).

#define DIM        64
#define NPAIR      2080                 // 64*65/2
#define KQUAD_BASE 64
#define KCUBE_BASE 2144                 // 64 + 2080
#define KTOT       135264               // 64 + 2080 + 64*2080
#define KSPLIT     4
#define MCUBE_PER_SPLIT (DIM / KSPLIT)  // 16
#define BATCH      1024

// LDS row strides: even (8B-aligned b64) and ≡4 mod 32 dwords -> 64-bank conflict-free fragments
#define XS_STRIDE  68
#define S_STRIDE   2084

typedef float v2f __attribute__((ext_vector_type(2)));
typedef float v8f __attribute__((ext_vector_type(8)));

__device__ __forceinline__ v8f wmma_f32(v2f a, v2f b, v8f c) {
    // (neg_a, A, neg_b, B, c_mod, C, reuse_a, reuse_b) -> v_wmma_f32_16x16x4_f32
    return __builtin_amdgcn_wmma_f32_16x16x4_f32(false, a, false, b, (short)0, c,
                                                 false, false);
}

// Run one contiguous K-segment of length len (multiple of 32).
// aptr: per-lane LDS pointer (row base + koff), bptr: per-lane W pointer (row base + kbase + koff).
template <bool MUL>
__device__ __forceinline__ void krun(const float* __restrict__ aptr,
                                     const float* __restrict__ bptr,
                                     int len, float mult, v8f& acc0, v8f& acc1) {
    for (int k = 0; k < len; k += 32) {
        __builtin_prefetch(bptr + k + 512, 0, 0);   // global_prefetch_b8, 2KB ahead in W stream
#pragma unroll
        for (int kk = k; kk < k + 32; kk += 8) {
            v2f a0 = *(const v2f*)(aptr + kk);
            v2f b0 = *(const v2f*)(bptr + kk);
            v2f a1 = *(const v2f*)(aptr + kk + 4);
            v2f b1 = *(const v2f*)(bptr + kk + 4);
            if (MUL) { a0 *= mult; a1 *= mult; }
            acc0 = wmma_f32(a0, b0, acc0);
            acc1 = wmma_f32(a1, b1, acc1);
        }
    }
}

// One layer: out_partial[ks][16-row tile][Npad cols] = phi(x_tile) @ W^T  (K-partitioned)
template <int NT>
__global__ __launch_bounds__(NT * 32) void cubic_layer_kernel(
        const float* __restrict__ x, const float* __restrict__ W,
        float* __restrict__ partial, int N, int Npad) {
    __shared__ float xs[16 * XS_STRIDE];   //  4,352 B
    __shared__ float sq[16 * S_STRIDE];    // 133,376 B  (total 137,728 B -> 2 WGs/WGP)

    const int tid   = threadIdx.x;
    const int lane  = tid & 31;
    const int wv    = tid >> 5;        // n-tile owned by this wave
    const int mtile = blockIdx.x;      // 0..63
    const int ks    = blockIdx.y;      // 0..KSPLIT-1
    const int r0    = mtile * 16;
    const int NTHREADS = NT * 32;

    // ---- stage x tile (16 x 64) into LDS (b128 stores) ----
    for (int i = tid; i < 16 * 16; i += NTHREADS) {
        int r = i >> 4, c4 = (i & 15) * 4;
        float4 v = *(const float4*)(x + (size_t)(r0 + r) * DIM + c4);
        *(float4*)(&xs[r * XS_STRIDE + c4]) = v;
    }
    __syncthreads();

    // ---- build s = triu(x ⊗ x) per row in LDS (jnp.triu_indices order) ----
    for (int q = tid; q < 16 * 4096; q += NTHREADS) {
        int r = q >> 12, ij = q & 4095;
        int i = ij >> 6, j = ij & 63;
        if (j >= i) {
            int p = i * DIM - (i * (i - 1)) / 2 + (j - i);
            sq[r * S_STRIDE + p] = xs[r * XS_STRIDE + i] * xs[r * XS_STRIDE + j];
        }
    }
    __syncthreads();

    // ---- per-lane fragment bases (ISA §7.12.2 layouts) ----
    const int rr   = lane & 15;          // A row / D col index
    const int koff = (lane >> 4) * 2;    // K pair selected by lane half
    int ncol = wv * 16 + (lane & 15);
    int neff = ncol < N ? ncol : (N - 1);            // clamp keeps loads in bounds
    const float* wrow   = W + (size_t)neff * KTOT + koff;
    const float* arow_x = xs + rr * XS_STRIDE + koff;
    const float* arow_s = sq + rr * S_STRIDE + koff;

    v8f acc0 = {}, acc1 = {};

    if (ks == 0) {                                   // linear + quadratic segments
        krun<false>(arow_x, wrow, DIM, 1.0f, acc0, acc1);
        krun<false>(arow_s, wrow + KQUAD_BASE, NPAIR, 1.0f, acc0, acc1);
    }
    const int m0 = ks * MCUBE_PER_SPLIT;             // cubic segment: x_m * s
    for (int m = m0; m < m0 + MCUBE_PER_SPLIT; ++m) {
        float xm = xs[rr * XS_STRIDE + m];
        krun<true>(arow_s, wrow + KCUBE_BASE + m * NPAIR, NPAIR, xm, acc0, acc1);
    }
    acc0 += acc1;

    // ---- D layout: lane%16 = col, VGPR v -> row (lane/16)*8 + v ----
    float* pout = partial + ((size_t)ks * BATCH + r0) * Npad + wv * 16 + (lane & 15);
    const int mbase = (lane >> 4) * 8;
#pragma unroll
    for (int v = 0; v < 8; ++v)
        pout[(size_t)(mbase + v) * Npad] = acc0[v];
}

__global__ void reduce_bias_kernel(const float* __restrict__ partial,
                                   const float* __restrict__ bias,
                                   float* __restrict__ out, int N, int Npad) {
    int idx = blockIdx.x * blockDim.x + threadIdx.x;
    if (idx >= BATCH * N) return;
    int b = idx / N, n = idx - b * N;
    float s = bias[n];
#pragma unroll
    for (int ks = 0; ks < KSPLIT; ++ks)
        s += partial[((size_t)ks * BATCH + b) * Npad + n];
    out[idx] = s;
}

extern "C" void kernel_launch(void* const* d_in, const int* in_sizes, int n_in,
                              void* d_out, int out_size, void* d_ws, size_t ws_size,
                              hipStream_t stream) {
    const float* x  = (const float*)d_in[0];
    const float* W0 = (const float*)d_in[1];
    const float* b0 = (const float*)d_in[2];
    const float* W1 = (const float*)d_in[3];
    const float* b1 = (const float*)d_in[4];
    const float* W2 = (const float*)d_in[5];
    const float* b2 = (const float*)d_in[6];
    float* out = (float*)d_out;

    // workspace: act1 | act2 | partial (KSPLIT x 1024 x 64)  -> 1.5 MB total
    float* act1    = (float*)d_ws;
    float* act2    = act1 + (size_t)BATCH * DIM;
    float* partial = act2 + (size_t)BATCH * DIM;

    dim3 grid(BATCH / 16, KSPLIT);
    int rblk64 = (BATCH * 64 + 255) / 256;
    int rblk10 = (BATCH * 10 + 255) / 256;

    cubic_layer_kernel<4><<<grid, 128, 0, stream>>>(x, W0, partial, 64, 64);
    reduce_bias_kernel<<<rblk64, 256, 0, stream>>>(partial, b0, act1, 64, 64);

    cubic_layer_kernel<4><<<grid, 128, 0, stream>>>(act1, W1, partial, 64, 64);
    reduce_bias_kernel<<<rblk64, 256, 0, stream>>>(partial, b1, act2, 64, 64);

    cubic_layer_kernel<1><<<grid, 32, 0, stream>>>(act2, W2, partial, 10, 16);
    reduce_bias_kernel<<<rblk10, 256, 0, stream>>>(partial, b2, out, 10, 16);
}